// MDL_RNN_mante_26645977104954
// MI455X (gfx1250) — compile-verified
//
#include <hip/hip_runtime.h>

// ---------------------------------------------------------------------------
// MI455X spiking-RNN forward (compile-only tuned).
//  Phase 1 (parallel, exact-f32 WMMA): F[t,b,h] = Win @ x[t,b,:]
//    19200 x 2048 x 128 GEMM, 2x2 (32x32) tiles per wave of
//    v_wmma_f32_16x16x4_f32 -> 1:1 load:wmma instruction ratio.
//  Phase 2 (sequential): 300-step scan with the rank-16 factorization
//    Wr@r = (l*pin) @ (pout^T @ r). Batches are independent, so no grid sync:
//    16 blocks x 1024 threads, 4 batches per block (4-way parameter reuse),
//    thread owns h = j*256 + (tid&255)  (all loads lane-contiguous/coalesced),
//    19-float (16 u + 3 y) reduction per step via wave32 shfl_xor + LDS.
// Roofline: ~10 GFLOP GEMM + ~3 GFLOP VALU scan; F traffic 2x157 MB at
// 23.3 TB/s HBM; params (280 KB) stay hot in L2 (192 MB) with coalesced reads.
// ---------------------------------------------------------------------------

typedef float v2f __attribute__((ext_vector_type(2)));
typedef float v4f __attribute__((ext_vector_type(4)));
typedef float v8f __attribute__((ext_vector_type(8)));

#define T_STEPS 300
#define B_SZ    64
#define IN_SZ   128
#define H_SZ    2048
#define O_SZ    3
#define P_SZ    16
#define GB      4        // batches per scan block

static constexpr float LS    = 0.90483741803596f;  // exp(-dt/taus)
static constexpr float LM    = 0.95122942450071f;  // exp(-dt/taum)
static constexpr float LDC   = 0.96721610048201f;  // exp(-dt/taud)
static constexpr float CR    = 0.033333333333333f; // dt/taud
static constexpr float TREF_ = 0.005f;
static constexpr float VTHR_ = 1.0f;
static constexpr float DT_   = 0.001f;

#if defined(__HIP_DEVICE_COMPILE__) && defined(__gfx1250__) && \
    __has_builtin(__builtin_amdgcn_wmma_f32_16x16x4_f32)
  #define USE_WMMA 1
#endif

// ---------------------------------------------------------------------------
// Phase 1: F[n][h] = sum_k Win[h][k] * X[n][k];  n = t_local*B + b.
// One wave computes a 32(M=h) x 32(N=n) macro-tile as 2x2 WMMA tiles.
// ---------------------------------------------------------------------------
__global__ __launch_bounds__(256)
void ff_gemm(const float* __restrict__ X,    // [Ntot,128] (chunk rows of x)
             const float* __restrict__ Win,  // [2048,128]
             float* __restrict__ F,          // [Ntot,2048]
             int Ntot)
{
  const int wave = threadIdx.x >> 5;
  const int lane = threadIdx.x & 31;
  const int tile = blockIdx.x * 8 + wave;
  const int nth  = H_SZ >> 5;                     // 64 macro-tiles along h
  const int ntiles = nth * (Ntot >> 5);
  if (tile >= ntiles) return;                     // wave-uniform: EXEC all-1s
  const int th2 = tile % nth;                     // waves in block share tn2
  const int tn2 = tile / nth;

  const int lo = lane & 15;
  const int hi = lane >> 4;
  // A 16x4 f32 frag (ISA 7.12.2): lanes0-15 K={0,1}, lanes16-31 K={2,3}.
  const float* wa0 = Win + (size_t)(th2 * 32 + lo) * IN_SZ + 2 * hi;
  const float* wa1 = wa0 + (size_t)16 * IN_SZ;
  const float* xb0 = X   + (size_t)(tn2 * 32 + lo) * IN_SZ + 2 * hi;
  const float* xb1 = xb0 + (size_t)16 * IN_SZ;

  v8f c00 = {}, c01 = {}, c10 = {}, c11 = {};

#if defined(USE_WMMA)
  #pragma unroll 4
  for (int kb = 0; kb < IN_SZ; kb += 4) {
    const v2f a0 = *(const v2f*)(wa0 + kb);
    const v2f a1 = *(const v2f*)(wa1 + kb);
    const v2f b0 = *(const v2f*)(xb0 + kb);
    const v2f b1 = *(const v2f*)(xb1 + kb);
    c00 = __builtin_amdgcn_wmma_f32_16x16x4_f32(false, a0, false, b0, (short)0, c00, false, false);
    c01 = __builtin_amdgcn_wmma_f32_16x16x4_f32(false, a0, false, b1, (short)0, c01, false, false);
    c10 = __builtin_amdgcn_wmma_f32_16x16x4_f32(false, a1, false, b0, (short)0, c10, false, false);
    c11 = __builtin_amdgcn_wmma_f32_16x16x4_f32(false, a1, false, b1, (short)0, c11, false, false);
  }
#else
  // host-parse / non-gfx1250 fallback (never executed on MI455X)
  for (int kb = 0; kb < IN_SZ; ++kb) {
    c00[0] += wa0[kb] * xb0[kb];  c01[0] += wa0[kb] * xb1[kb];
    c10[0] += wa1[kb] * xb0[kb];  c11[0] += wa1[kb] * xb1[kb];
  }
#endif

  // D layout: VGPR j -> (M = j + 8*hi, N = lane&15): 8 contiguous h per lane.
  auto store_tile = [&](const v8f& c, int mBase, int nCol) {
    float* out = F + (size_t)nCol * H_SZ + mBase + 8 * hi;
    const float* cf = (const float*)&c;
    *(v4f*)(out)     = *(const v4f*)(cf);
    *(v4f*)(out + 4) = *(const v4f*)(cf + 4);
  };
  const int m0 = th2 * 32, n0 = tn2 * 32 + lo;
  store_tile(c00, m0,      n0);
  store_tile(c01, m0,      n0 + 16);
  store_tile(c10, m0 + 16, n0);
  store_tile(c11, m0 + 16, n0 + 16);
}

// ---------------------------------------------------------------------------
// Phase 2: per-batch scan. Block handles GB=4 batches; sub-group sg (256 thr)
// owns batch b = blockIdx*GB+sg; thread owns h = j*256 + (tid&255), j=0..7.
// All global accesses are lane-contiguous; 4 sub-groups stream the same
// parameter range in lockstep (WGP$/L2 reuse).
// ---------------------------------------------------------------------------
__global__ __launch_bounds__(256 * GB)
void scan_chunk(const float* __restrict__ F,     // [tc*B, 2048] this chunk
                const float* __restrict__ pin,   // [2048,16]
                const float* __restrict__ pout,  // [2048,16]
                const float* __restrict__ lvec,  // [16]
                const float* __restrict__ Wout,  // [3,2048]
                float* __restrict__ Y,           // [300,64,3]
                float* __restrict__ st,          // state in ws
                int t0, int tc, int first)
{
  const int tid  = threadIdx.x;
  const int sg   = tid >> 8;        // sub-group = batch within block
  const int lt   = tid & 255;       // thread within sub-group
  const int b    = blockIdx.x * GB + sg;
  const int wv   = (tid >> 5) & 7;  // wave within sub-group
  const int lane = tid & 31;

  float* stI = st;
  float* stM = st + (size_t)1 * B_SZ * H_SZ;
  float* stS = st + (size_t)2 * B_SZ * H_SZ;
  float* stR = st + (size_t)3 * B_SZ * H_SZ;
  float* stT = st + (size_t)4 * B_SZ * H_SZ;
  float* stU = st + (size_t)5 * B_SZ * H_SZ;     // [B,16]
  const size_t sb = (size_t)b * H_SZ + lt;       // + j*256 per element

  __shared__ float red[GB][8][20];
  __shared__ float uq[GB][16];
  __shared__ float lsh[16];

  float I[8], mem[8], s[8], r[8], tl[8];
  if (first) {
    #pragma unroll
    for (int j = 0; j < 8; ++j) { I[j]=0.f; mem[j]=0.f; s[j]=0.f; r[j]=0.f; tl[j]=-1.f; }
    if (lt < 16) uq[sg][lt] = 0.f;
  } else {
    #pragma unroll
    for (int j = 0; j < 8; ++j) {
      const size_t ix = sb + (size_t)j * 256;
      I[j]=stI[ix]; mem[j]=stM[ix]; s[j]=stS[ix]; r[j]=stR[ix]; tl[j]=stT[ix];
    }
    if (lt < 16) uq[sg][lt] = stU[b * 16 + lt];
  }
  if (tid < 16) lsh[tid] = lvec[tid];
  __syncthreads();

  for (int k = 0; k < tc; ++k) {
    const float ft = DT_ * (float)(t0 + k);
    float lu[16];
    #pragma unroll
    for (int p = 0; p < 16; ++p) lu[p] = lsh[p] * uq[sg][p];

    const float* Fr = F + ((size_t)k * B_SZ + b) * H_SZ + lt;

    float part[19];
    #pragma unroll
    for (int p = 0; p < 19; ++p) part[p] = 0.f;

    #pragma unroll
    for (int j = 0; j < 8; ++j) {
      const int h = j * 256 + lt;
      // I = ls*I + F + (l*pin)[h,:] . u        (rank-16 recurrent drive)
      const float* pr = pin + (size_t)h * 16;
      float acc = 0.f;
      #pragma unroll
      for (int p = 0; p < 16; p += 4) {
        const v4f pv = *(const v4f*)(pr + p);
        acc += pv[0]*lu[p] + pv[1]*lu[p+1] + pv[2]*lu[p+2] + pv[3]*lu[p+3];
      }
      I[j] = LS * I[j] + Fr[(size_t)j * 256] + acc;
      // membrane with refractory mask + reset-by-spike (previous s)
      const float mask = (ft > tl[j] + TREF_) ? 1.f : 0.f;
      mem[j] = mask * (LM * mem[j] + (1.f - LM) * I[j]) * (1.f - s[j]);
      // synaptic filter uses previous spike, then new spike + tlast
      r[j]  = LDC * r[j] + CR * s[j];
      s[j]  = (mem[j] > VTHR_) ? 1.f : 0.f;
      tl[j] += (ft - tl[j]) * s[j];
      // partials: u_next[p] = sum_h r*pout[h,p]; y[o] = sum_h Wout[o,h]*r
      const float* qr = pout + (size_t)h * 16;
      #pragma unroll
      for (int p = 0; p < 16; p += 4) {
        const v4f qv = *(const v4f*)(qr + p);
        part[p]   += r[j] * qv[0];
        part[p+1] += r[j] * qv[1];
        part[p+2] += r[j] * qv[2];
        part[p+3] += r[j] * qv[3];
      }
      part[16] += Wout[h]            * r[j];
      part[17] += Wout[H_SZ + h]     * r[j];
      part[18] += Wout[2 * H_SZ + h] * r[j];
    }

    // wave32 butterfly reduce, then 8-wave combine per sub-group in LDS
    #pragma unroll
    for (int p = 0; p < 19; ++p) {
      float v = part[p];
      v += __shfl_xor(v, 1, 32);
      v += __shfl_xor(v, 2, 32);
      v += __shfl_xor(v, 4, 32);
      v += __shfl_xor(v, 8, 32);
      v += __shfl_xor(v, 16, 32);
      part[p] = v;
    }
    if (lane == 0) {
      #pragma unroll
      for (int p = 0; p < 19; ++p) red[sg][wv][p] = part[p];
    }
    __syncthreads();
    if (lt < 19) {
      float v = red[sg][0][lt] + red[sg][1][lt] + red[sg][2][lt] + red[sg][3][lt]
              + red[sg][4][lt] + red[sg][5][lt] + red[sg][6][lt] + red[sg][7][lt];
      if (lt < 16) uq[sg][lt] = v;
      else Y[((size_t)(t0 + k) * B_SZ + b) * O_SZ + (lt - 16)] = v;
    }
    __syncthreads();
  }

  #pragma unroll
  for (int j = 0; j < 8; ++j) {
    const size_t ix = sb + (size_t)j * 256;
    stI[ix]=I[j]; stM[ix]=mem[j]; stS[ix]=s[j]; stR[ix]=r[j]; stT[ix]=tl[j];
  }
  if (lt < 16) stU[b * 16 + lt] = uq[sg][lt];
}

// ---------------------------------------------------------------------------
extern "C" void kernel_launch(void* const* d_in, const int* in_sizes, int n_in,
                              void* d_out, int out_size, void* d_ws, size_t ws_size,
                              hipStream_t stream) {
  const float* x    = (const float*)d_in[0];  // [300,64,128,1]
  const float* Win  = (const float*)d_in[1];  // [2048,128]
  const float* Wout = (const float*)d_in[2];  // [3,2048]
  const float* pin  = (const float*)d_in[3];  // [2048,16]
  const float* pout = (const float*)d_in[4];  // [2048,16]
  const float* lvec = (const float*)d_in[5];  // [16]
  float* y  = (float*)d_out;                  // [300,64,3,1]
  float* ws = (float*)d_ws;

  // ws layout: [I|mem|s|r|tlast] 5*B*H + u B*16, then F chunk buffer.
  const size_t stateFloats = (size_t)5 * B_SZ * H_SZ + (size_t)B_SZ * 16;
  const size_t fOff        = (stateFloats + 63) & ~(size_t)63;
  float* F = ws + fOff;

  long long avail = (long long)(ws_size / 4) - (long long)fOff;
  long long tcap  = avail / ((long long)B_SZ * H_SZ);
  int TC = (int)(tcap < 1 ? 1 : (tcap > T_STEPS ? T_STEPS : tcap));

  for (int t0 = 0; t0 < T_STEPS; t0 += TC) {
    const int tc   = (T_STEPS - t0 < TC) ? (T_STEPS - t0) : TC;
    const int Ntot = tc * B_SZ;
    const int tiles  = (H_SZ / 32) * (Ntot / 32);
    const int blocks = (tiles + 7) / 8;
    ff_gemm<<<blocks, 256, 0, stream>>>(x + (size_t)t0 * B_SZ * IN_SZ, Win, F, Ntot);
    scan_chunk<<<B_SZ / GB, 256 * GB, 0, stream>>>(F, pin, pout, lvec, Wout, y, ws,
                                                   t0, tc, t0 == 0 ? 1 : 0);
  }
}